// GaussianUnit_27986006901036
// MI455X (gfx1250) — compile-verified
//
#include <hip/hip_runtime.h>
#include <math.h>

#define R_   5
#define K_   10
#define C_   384
#define RK   50
#define RKP  64   // padded rk for WMMA tiles (4 x 16)

typedef float v2f __attribute__((ext_vector_type(2)));
typedef float v8f __attribute__((ext_vector_type(8)));
typedef int   v4i __attribute__((ext_vector_type(4)));

// async b128 builtin wants (global v4i*, lds v4i*, imm offset, imm cpol)
#define GPTR128(p) ((__attribute__((address_space(1))) v4i*)(p))
#define LPTR128(p) ((__attribute__((address_space(3))) v4i*)(p))

// ---------------------------------------------------------------- binning ---
__global__ void init_counts_kernel(int* counts, int G) {
  int i = blockIdx.x * blockDim.x + threadIdx.x;
  if (i < G) counts[i] = 0;
}

__global__ void hist_kernel(const int* __restrict__ gid, int* counts, int N) {
  int i = blockIdx.x * blockDim.x + threadIdx.x;
  if (i < N) atomicAdd(&counts[gid[i]], 1);
}

__global__ void scan_kernel(const int* __restrict__ counts, int* offs, int* cursor, int G) {
  if (threadIdx.x == 0) {
    int acc = 0;
    for (int g = 0; g < G; ++g) { offs[g] = acc; cursor[g] = acc; acc += counts[g]; }
    offs[G] = acc;
  }
}

__global__ void scatter_kernel(const int* __restrict__ gid, int* cursor, int* order, int N) {
  int i = blockIdx.x * blockDim.x + threadIdx.x;
  if (i < N) { int p = atomicAdd(&cursor[gid[i]], 1); order[p] = i; }
}

// ------------------------------------------------- per-neighborhood setup ---
// Winv[g] = kron(LTinv, LSinv); store B[g][j][i] = Winv[i][j] padded to 64x64
__global__ __launch_bounds__(64) void precompute_kernel(
    const float* __restrict__ T, const float* __restrict__ S,
    const float* __restrict__ mean_full, const int* __restrict__ nbh,
    float* __restrict__ Bws, float* __restrict__ nu_g, float* __restrict__ mu_g,
    float* __restrict__ nn_g)
{
  const int g = blockIdx.x;
  const int tid = threadIdx.x;
  __shared__ float sLT[R_ * R_];
  __shared__ float sLS[K_ * K_];
  __shared__ float sMu[RKP];
  __shared__ float sNu[RKP];
  __shared__ int   sNb[K_];

  if (tid < K_) sNb[tid] = nbh[g * K_ + tid];
  __syncthreads();

  if (tid == 0) {                       // chol(T) 5x5 then triangular inverse
    float A[R_][R_], Lm[R_][R_], Li[R_][R_];
    for (int i = 0; i < R_; ++i)
      for (int j = 0; j < R_; ++j) { A[i][j] = T[i * R_ + j]; Lm[i][j] = 0.f; Li[i][j] = 0.f; }
    for (int i = 0; i < R_; ++i)
      for (int j = 0; j <= i; ++j) {
        float s = A[i][j];
        for (int p = 0; p < j; ++p) s -= Lm[i][p] * Lm[j][p];
        Lm[i][j] = (i == j) ? sqrtf(s) : s / Lm[j][j];
      }
    for (int c = 0; c < R_; ++c) {
      Li[c][c] = 1.0f / Lm[c][c];
      for (int r = c + 1; r < R_; ++r) {
        float s = 0.f;
        for (int p = c; p < r; ++p) s += Lm[r][p] * Li[p][c];
        Li[r][c] = -s / Lm[r][r];
      }
    }
    for (int i = 0; i < R_; ++i)
      for (int j = 0; j < R_; ++j) sLT[i * R_ + j] = Li[i][j];
  } else if (tid == 1) {                // chol(Sg) 10x10 then triangular inverse
    float A[K_][K_], Lm[K_][K_], Li[K_][K_];
    for (int i = 0; i < K_; ++i)
      for (int j = 0; j < K_; ++j) { A[i][j] = S[sNb[i] * C_ + sNb[j]]; Lm[i][j] = 0.f; Li[i][j] = 0.f; }
    for (int i = 0; i < K_; ++i)
      for (int j = 0; j <= i; ++j) {
        float s = A[i][j];
        for (int p = 0; p < j; ++p) s -= Lm[i][p] * Lm[j][p];
        Lm[i][j] = (i == j) ? sqrtf(s) : s / Lm[j][j];
      }
    for (int c = 0; c < K_; ++c) {
      Li[c][c] = 1.0f / Lm[c][c];
      for (int r = c + 1; r < K_; ++r) {
        float s = 0.f;
        for (int p = c; p < r; ++p) s += Lm[r][p] * Li[p][c];
        Li[r][c] = -s / Lm[r][r];
      }
    }
    for (int i = 0; i < K_; ++i)
      for (int j = 0; j < K_; ++j) sLS[i * K_ + j] = Li[i][j];
  }
  __syncthreads();

  for (int j = tid; j < RKP; j += 64) {       // gathered neighborhood means
    float v = 0.f;
    if (j < RK) { int r = j / K_, k = j % K_; v = mean_full[r * C_ + sNb[k]]; }
    sMu[j] = v;
  }
  __syncthreads();

  for (int i = tid; i < RKP; i += 64) {       // nu_g = kron(LT,LS) @ mu
    float v = 0.f;
    if (i < RK) {
      int r1 = i / K_, k1 = i % K_;
      for (int r2 = 0; r2 < R_; ++r2) {
        float inner = 0.f;
        for (int k2 = 0; k2 < K_; ++k2) inner += sLS[k1 * K_ + k2] * sMu[r2 * K_ + k2];
        v += sLT[r1 * R_ + r2] * inner;
      }
    }
    sNu[i] = v;
    nu_g[g * RKP + i] = v;
    mu_g[g * RKP + i] = sMu[i];
  }
  __syncthreads();
  if (tid == 0) {
    float s = 0.f;
    for (int i = 0; i < RK; ++i) s += sNu[i] * sNu[i];
    nn_g[g] = s;
  }
  float* Bg = Bws + (size_t)g * RKP * RKP;    // B[j][i] = Winv[i][j], zero-padded
  for (int idx = tid; idx < RKP * RKP; idx += 64) {
    int j = idx >> 6, i = idx & 63;
    float v = 0.f;
    if (i < RK && j < RK)
      v = sLT[(i / K_) * R_ + (j / K_)] * sLS[(i % K_) * K_ + (j % K_)];
    Bg[idx] = v;
  }
}

// --------------------------------------------------- WMMA whitening GEMM ---
// 4 waves / block; 64 spikes share one async-staged B-matrix in LDS; each
// wave runs a 16x64x64 fp32 WMMA GEMM on its 16-spike sub-tile.
__global__ __launch_bounds__(128) void whiten_kernel(
    const float* __restrict__ features, const float* __restrict__ weights,
    const int* __restrict__ gid, const int* __restrict__ order,
    const float* __restrict__ Bws, const float* __restrict__ nu_g,
    const float* __restrict__ mu_g, const float* __restrict__ nn_g,
    float* __restrict__ ztn, float* __restrict__ ntn,
    float* __restrict__ zw_out, float* __restrict__ nu_out, int N)
{
  const int tid  = threadIdx.x;
  const int lane = tid & 31;
  const int wv   = tid >> 5;          // wave 0..3
  const int t0   = blockIdx.x * 64;   // first position of this block's span

  __shared__ float sB[RKP * RKP];     // 16 KB: Winv^T (K x N layout), padded
  __shared__ float sZ[4][16][RKP];    // 16 KB: per-wave spike tiles / ZW
  __shared__ float sNu[RKP];
  __shared__ float sMuS[RKP];
  __shared__ int   sIdx[4][16];

  const int last   = min(t0 + 63, N - 1);
  const int gfirst = gid[order[t0]];
  const int glast  = gid[order[last]];
  const bool uniform = (gfirst == glast) && (t0 + 63 < N);

  if (uniform) {
    const int g = gfirst;

    // 1) kick off async B copy: 1024 x b128, ASYNCcnt-tracked, overlaps gather
    {
      const float* Bg = Bws + (size_t)g * RKP * RKP;
      for (int idx = tid; idx < (RKP * RKP) / 4; idx += 128) {
        __builtin_amdgcn_global_load_async_to_lds_b128(
            GPTR128(Bg + 4 * idx), LPTR128(&sB[4 * idx]), 0, 0);
      }
    }

    // 2) meanwhile: stage nu/mu, spike ids, and build Z tiles in LDS
    for (int idx = tid; idx < RKP; idx += 128) {
      sNu[idx]  = nu_g[g * RKP + idx];
      sMuS[idx] = mu_g[g * RKP + idx];
    }
    if (lane < 16) sIdx[wv][lane] = order[t0 + wv * 16 + lane];
    __syncthreads();
    for (int idx = lane; idx < 16 * RKP; idx += 32) {   // Z = features - mu_g
      int m = idx >> 6, j = idx & 63;
      float v = 0.f;
      if (j < RK) v = features[(size_t)sIdx[wv][m] * RK + j] - sMuS[j];
      sZ[wv][m][j] = v;
    }
    __builtin_amdgcn_s_wait_asynccnt(0);   // our async B chunks have landed
    __syncthreads();                        // everyone's chunks visible

    // 3) ZW(16x64) = Z(16x64) @ B(64x64) via V_WMMA_F32_16X16X4_F32
    v8f acc0 = {}, acc1 = {}, acc2 = {}, acc3 = {};
    const int M = lane & 15;
    const int half = lane >> 4;
    #pragma unroll
    for (int k0 = 0; k0 < RKP; k0 += 4) {
      const int ka = k0 + 2 * half;
      v2f a; a.x = sZ[wv][M][ka]; a.y = sZ[wv][M][ka + 1];
      v2f b0, b1, b2, b3;
      b0.x = sB[ka * RKP +  0 + M]; b0.y = sB[(ka + 1) * RKP +  0 + M];
      b1.x = sB[ka * RKP + 16 + M]; b1.y = sB[(ka + 1) * RKP + 16 + M];
      b2.x = sB[ka * RKP + 32 + M]; b2.y = sB[(ka + 1) * RKP + 32 + M];
      b3.x = sB[ka * RKP + 48 + M]; b3.y = sB[(ka + 1) * RKP + 48 + M];
      acc0 = __builtin_amdgcn_wmma_f32_16x16x4_f32(false, a, false, b0, (short)0, acc0, false, false);
      acc1 = __builtin_amdgcn_wmma_f32_16x16x4_f32(false, a, false, b1, (short)0, acc1, false, false);
      acc2 = __builtin_amdgcn_wmma_f32_16x16x4_f32(false, a, false, b2, (short)0, acc2, false, false);
      acc3 = __builtin_amdgcn_wmma_f32_16x16x4_f32(false, a, false, b3, (short)0, acc3, false, false);
    }
    __syncthreads();
    #pragma unroll
    for (int v = 0; v < 8; ++v) {     // C/D layout: row = v + 8*(lane/16), col = 16*t + lane%16
      int row = v + 8 * half;
      sZ[wv][row][ 0 + M] = acc0[v];
      sZ[wv][row][16 + M] = acc1[v];
      sZ[wv][row][32 + M] = acc2[v];
      sZ[wv][row][48 + M] = acc3[v];
    }
    __syncthreads();
    for (int idx = lane; idx < 16 * RK; idx += 32) {
      int m = idx / RK, j = idx % RK;
      size_t base = (size_t)sIdx[wv][m] * RK + j;
      zw_out[base] = sZ[wv][m][j];
      nu_out[base] = sNu[j];
    }
    if (lane < 16) {                  // per-spike sufficient statistics for MAP scale
      int n = sIdx[wv][lane];
      float dot = 0.f;
      for (int j = 0; j < RK; ++j) dot += sZ[wv][lane][j] * sNu[j];
      float w = weights[n];
      float zn = w * dot;
      ztn[n] = zn * zn;
      ntn[n] = w * w * nn_g[g];
    }
  } else {                            // rare neighborhood-boundary span: scalar path
    float* sZflat = &sZ[0][0][0];     // 64 rows of RKP scratch
    int pos = t0 + tid;
    if (tid < 64 && pos < N) {
      int n = order[pos];
      int g = gid[n];
      const float* Bg  = Bws + (size_t)g * RKP * RKP;
      const float* nug = nu_g + (size_t)g * RKP;
      const float* mug = mu_g + (size_t)g * RKP;
      float* zr = sZflat + tid * RKP;
      for (int j = 0; j < RK; ++j) zr[j] = features[(size_t)n * RK + j] - mug[j];
      float dot = 0.f;
      for (int i = 0; i < RK; ++i) {
        float s = 0.f;
        for (int j = 0; j < RK; ++j) s += Bg[j * RKP + i] * zr[j];
        zw_out[(size_t)n * RK + i] = s;
        nu_out[(size_t)n * RK + i] = nug[i];
        dot += s * nug[i];
      }
      float w = weights[n];
      ztn[n] = (w * dot) * (w * dot);
      ntn[n] = w * w * nn_g[g];
    }
  }
}

// ------------------------------------------------ scalar Adam (MAP scale) ---
__global__ __launch_bounds__(1024) void scale_kernel(
    const float* __restrict__ ztn, const float* __restrict__ ntn,
    float* __restrict__ out_scale, int N)
{
  __shared__ float red1[32], red2[32];
  __shared__ float sT;
  __shared__ int sDone;
  const int tid = threadIdx.x;
  const int lane = tid & 31;
  const int wid = tid >> 5;
  if (tid == 0) { sT = 0.f; sDone = 0; }
  __syncthreads();
  const float a = 10.0f + 0.5f * (float)N - 1.0f;   // ALPHA + n/2 - 1
  const float beta = 100.0f, lr = 0.1f;
  const float bb1 = 0.9f, bb2 = 0.999f, eps = 1e-8f, xtol = 0.01f;
  float m = 0.f, v = 0.f, b1p = 1.f, b2p = 1.f;
  for (int j = 0; j < 2000; ++j) {
    b1p *= bb1; b2p *= bb2;
    float t = sT;
    float lam = expf(t);
    float s1 = 0.f, s2 = 0.f;                       // sum 1/npl, sum q/npl^2
    for (int n = tid; n < N; n += 1024) {
      float inv = 1.0f / (lam + ntn[n]);
      s1 += inv;
      s2 += ztn[n] * inv * inv;
    }
    for (int off = 16; off; off >>= 1) { s1 += __shfl_down(s1, off); s2 += __shfl_down(s2, off); }
    if (lane == 0) { red1[wid] = s1; red2[wid] = s2; }
    __syncthreads();
    if (tid == 0) {
      float S1 = 0.f, S2 = 0.f;
      for (int w2 = 0; w2 < 32; ++w2) { S1 += red1[w2]; S2 += red2[w2]; }
      float g = -(a - beta * lam - 0.5f * lam * S1 - 0.5f * lam * S2);
      m = bb1 * m + (1.f - bb1) * g;
      v = bb2 * v + (1.f - bb2) * g * g;
      float mhat = m / (1.f - b1p);
      float vhat = v / (1.f - b2p);
      float t1 = t - lr * mhat / (sqrtf(vhat) + eps);
      float lam1 = expf(t1);
      sT = t1;                                      // update commits on triggering iter
      if ((j > 8) && (fabsf(lam1 - lam) < xtol)) sDone = 1;
    }
    __syncthreads();
    if (sDone) break;
  }
  if (tid == 0) out_scale[0] = expf(-0.5f * sT);    // 1/sqrt(exp(t))
}

// -------------------------------------------------------------- launcher ---
extern "C" void kernel_launch(void* const* d_in, const int* in_sizes, int n_in,
                              void* d_out, int out_size, void* d_ws, size_t ws_size,
                              hipStream_t stream)
{
  const float* features  = (const float*)d_in[0];
  const float* mean_full = (const float*)d_in[1];
  const float* T         = (const float*)d_in[2];
  const float* S         = (const float*)d_in[3];
  const float* weights   = (const float*)d_in[4];
  const int*   gid       = (const int*)d_in[5];
  const int*   nbh       = (const int*)d_in[6];
  const int N = in_sizes[0] / RK;
  const int G = in_sizes[6] / K_;

  float* ws   = (float*)d_ws;
  float* Bws  = ws;                              // G * 64 * 64
  float* nu_g = Bws + (size_t)G * RKP * RKP;     // G * 64
  float* mu_g = nu_g + (size_t)G * RKP;          // G * 64
  float* nn_g = mu_g + (size_t)G * RKP;          // G
  float* ztn  = nn_g + G;                        // N
  float* ntn  = ztn + N;                         // N
  int* counts = (int*)(ntn + N);                 // G
  int* offs   = counts + G;                      // G + 1
  int* cursor = offs + G + 1;                    // G
  int* order  = cursor + G;                      // N

  float* zw_out    = (float*)d_out;
  float* nu_out    = zw_out + (size_t)N * RK;
  float* scale_out = nu_out + (size_t)N * RK;

  init_counts_kernel<<<(G + 63) / 64, 64, 0, stream>>>(counts, G);
  hist_kernel<<<(N + 255) / 256, 256, 0, stream>>>(gid, counts, N);
  scan_kernel<<<1, 32, 0, stream>>>(counts, offs, cursor, G);
  scatter_kernel<<<(N + 255) / 256, 256, 0, stream>>>(gid, cursor, order, N);
  precompute_kernel<<<G, 64, 0, stream>>>(T, S, mean_full, nbh, Bws, nu_g, mu_g, nn_g);
  whiten_kernel<<<(N + 63) / 64, 128, 0, stream>>>(features, weights, gid, order,
                                                   Bws, nu_g, mu_g, nn_g,
                                                   ztn, ntn, zw_out, nu_out, N);
  scale_kernel<<<1, 1024, 0, stream>>>(ztn, ntn, scale_out, N);
}